// ManualGATLayer_90391881712253
// MI455X (gfx1250) — compile-verified
//
#include <hip/hip_runtime.h>
#include <math.h>

typedef __attribute__((ext_vector_type(2))) float v2f;
typedef __attribute__((ext_vector_type(8))) float v8f;

__device__ __forceinline__ int float_to_ordered_int(float f) {
    int i = __float_as_int(f);
    return (i >= 0) ? i : (i ^ 0x7FFFFFFF);
}
__device__ __forceinline__ float ordered_int_to_float(int i) {
    return __int_as_float((i >= 0) ? i : (i ^ 0x7FFFFFFF));
}

// ---------------- K0: zero accumulators, init global max ----------------
__global__ void k_init(float* __restrict__ out_nodes, int n_out,
                       float* __restrict__ alpha_sum, int n_nodes,
                       int* __restrict__ gmax) {
    int i = blockIdx.x * blockDim.x + threadIdx.x;
    if (i < n_out)   out_nodes[i] = 0.0f;
    if (i < n_nodes) alpha_sum[i] = 0.0f;
    if (i == 0)      *gmax = float_to_ordered_int(-INFINITY);
}

// ---------------- K1: Wh = h @ W^T via V_WMMA_F32_16X16X4_F32 ----------------
// One wave computes a 16(M) x 64(N) tile of Wh. A = h[16x4], B = W^T[4x16].
__global__ void __launch_bounds__(256) k_gemm_wmma(const float* __restrict__ h,
                                                   const float* __restrict__ W,
                                                   float* __restrict__ Wh,
                                                   int n_nodes) {
    const int wave = (int)((blockIdx.x * blockDim.x + threadIdx.x) >> 5);
    const int lane = threadIdx.x & 31;
    const int m0   = wave * 16;
    if (m0 >= n_nodes) return;            // wave-uniform: EXEC stays all-ones

    const int half = lane >> 4;           // 0: lanes 0-15, 1: lanes 16-31
    const int l16  = lane & 15;
    const int koff = half * 2;            // K sub-offset within the 16x4 frag

    v8f c0 = {}, c1 = {}, c2 = {}, c3 = {};
    const float* arow = h + (size_t)(m0 + l16) * 128 + koff;
    const float* b0p  = W + (size_t)(0  + l16) * 128 + koff;
    const float* b1p  = W + (size_t)(16 + l16) * 128 + koff;
    const float* b2p  = W + (size_t)(32 + l16) * 128 + koff;
    const float* b3p  = W + (size_t)(48 + l16) * 128 + koff;

    #pragma unroll 4
    for (int k0 = 0; k0 < 128; k0 += 4) {
        v2f a  = *(const v2f*)(arow + k0);
        v2f b0 = *(const v2f*)(b0p + k0);
        v2f b1 = *(const v2f*)(b1p + k0);
        v2f b2 = *(const v2f*)(b2p + k0);
        v2f b3 = *(const v2f*)(b3p + k0);
        c0 = __builtin_amdgcn_wmma_f32_16x16x4_f32(false, a, false, b0, (short)0, c0, false, false);
        c1 = __builtin_amdgcn_wmma_f32_16x16x4_f32(false, a, false, b1, (short)0, c1, false, false);
        c2 = __builtin_amdgcn_wmma_f32_16x16x4_f32(false, a, false, b2, (short)0, c2, false, false);
        c3 = __builtin_amdgcn_wmma_f32_16x16x4_f32(false, a, false, b3, (short)0, c3, false, false);
    }

    // D layout: lanes 0-15 -> (M=v, N=l16); lanes 16-31 -> (M=8+v, N=l16)
    float* o = Wh + (size_t)(m0 + half * 8) * 64 + l16;
    #pragma unroll
    for (int v = 0; v < 8; ++v) {
        o[(size_t)v * 64 + 0]  = c0[v];
        o[(size_t)v * 64 + 16] = c1[v];
        o[(size_t)v * 64 + 32] = c2[v];
        o[(size_t)v * 64 + 48] = c3[v];
    }
}

// ---------------- K2: s1[n] = Wh[n]·a_w[0:64], s2[n] = Wh[n]·a_w[64:128] ----------------
__global__ void __launch_bounds__(256) k_attn_scalars(const float* __restrict__ Wh,
                                                      const float* __restrict__ a_w,
                                                      float* __restrict__ s1,
                                                      float* __restrict__ s2,
                                                      int n_nodes) {
    const int wave = (int)((blockIdx.x * blockDim.x + threadIdx.x) >> 5);
    const int lane = threadIdx.x & 31;
    if (wave >= n_nodes) return;
    const float* row = Wh + (size_t)wave * 64;
    float w_lo = row[lane], w_hi = row[lane + 32];
    float p1 = w_lo * a_w[lane]      + w_hi * a_w[lane + 32];
    float p2 = w_lo * a_w[64 + lane] + w_hi * a_w[96 + lane];
    #pragma unroll
    for (int off = 16; off > 0; off >>= 1) {
        p1 += __shfl_xor(p1, off, 32);
        p2 += __shfl_xor(p2, off, 32);
    }
    if (lane == 0) { s1[wave] = p1; s2[wave] = p2; }
}

// ---------------- K3: alpha = leaky_relu(s1[src]+s2[dst]); global max ----------------
__global__ void k_alpha_max(const int* __restrict__ src, const int* __restrict__ dst,
                            const float* __restrict__ s1, const float* __restrict__ s2,
                            float* __restrict__ alpha_io, int* __restrict__ gmax,
                            int n_edges) {
    int e = blockIdx.x * blockDim.x + threadIdx.x;
    float a = -INFINITY;
    if (e < n_edges) {
        float x = s1[src[e]] + s2[dst[e]];
        a = x > 0.0f ? x : 0.2f * x;
        alpha_io[e] = a;
    }
    #pragma unroll
    for (int off = 16; off > 0; off >>= 1) a = fmaxf(a, __shfl_xor(a, off, 32));
    if ((threadIdx.x & 31) == 0) atomicMax(gmax, float_to_ordered_int(a));
}

// ---------------- K4: alpha_exp = exp(alpha - max); segment-sum over dst ----------------
__global__ void k_exp_sum(const int* __restrict__ dst, float* __restrict__ alpha_io,
                          float* __restrict__ alpha_sum, const int* __restrict__ gmax,
                          int n_edges) {
    int e = blockIdx.x * blockDim.x + threadIdx.x;
    if (e >= n_edges) return;
    float M = ordered_int_to_float(*gmax);
    float ex = expf(alpha_io[e] - M);
    alpha_io[e] = ex;
    unsafeAtomicAdd(&alpha_sum[dst[e]], ex);
}

// ---------------- K5: normalize + weighted scatter-add (one wave per edge) ----------------
__global__ void __launch_bounds__(256) k_scatter(const int* __restrict__ src,
                                                 const int* __restrict__ dst,
                                                 const float* __restrict__ Wh,
                                                 const float* __restrict__ alpha_sum,
                                                 float* __restrict__ alpha_io,
                                                 float* __restrict__ out_nodes,
                                                 int n_edges) {
    const long long gtid = (long long)blockIdx.x * blockDim.x + threadIdx.x;
    const int wave = (int)(gtid >> 5);
    const int lane = threadIdx.x & 31;
    if (wave >= n_edges) return;
    const int s = src[wave], d = dst[wave];
    const float norm = alpha_io[wave] / (alpha_sum[d] + 1e-9f);
    if (lane == 0) alpha_io[wave] = norm;   // final alpha_norm output
    const float* wrow = Wh + (size_t)s * 64;
    float* orow = out_nodes + (size_t)d * 64;
    unsafeAtomicAdd(&orow[lane],      norm * wrow[lane]);
    unsafeAtomicAdd(&orow[lane + 32], norm * wrow[lane + 32]);
}

// ---------------- K6: ELU ----------------
__global__ void k_elu(float* __restrict__ out_nodes, int n) {
    int i = blockIdx.x * blockDim.x + threadIdx.x;
    if (i >= n) return;
    float x = out_nodes[i];
    out_nodes[i] = x > 0.0f ? x : expm1f(x);
}

extern "C" void kernel_launch(void* const* d_in, const int* in_sizes, int n_in,
                              void* d_out, int out_size, void* d_ws, size_t ws_size,
                              hipStream_t stream) {
    const float* h   = (const float*)d_in[0];   // [N, 128]
    const float* W   = (const float*)d_in[1];   // [64, 128]
    const float* a_w = (const float*)d_in[2];   // [1, 128]
    const int*   ei  = (const int*)d_in[3];     // [2, E]

    const int N = in_sizes[0] / 128;
    const int E = in_sizes[3] / 2;
    const int* src = ei;
    const int* dst = ei + E;

    float* out_nodes = (float*)d_out;                       // N*64 floats
    float* alpha_io  = (float*)d_out + (size_t)N * 64;      // E floats (alpha scratch -> alpha_norm)

    char* ws = (char*)d_ws;
    float* Wh        = (float*)ws;  ws += (size_t)N * 64 * sizeof(float);
    float* s1        = (float*)ws;  ws += (size_t)N * sizeof(float);
    float* s2        = (float*)ws;  ws += (size_t)N * sizeof(float);
    float* alpha_sum = (float*)ws;  ws += (size_t)N * sizeof(float);
    int*   gmax      = (int*)ws;

    const int n_out_elems = N * 64;

    k_init<<<(n_out_elems + 255) / 256, 256, 0, stream>>>(out_nodes, n_out_elems, alpha_sum, N, gmax);

    {
        int mtiles = (N + 15) / 16;        // waves needed
        int blocks = (mtiles + 7) / 8;     // 8 waves per 256-thread block
        k_gemm_wmma<<<blocks, 256, 0, stream>>>(h, W, Wh, N);
    }

    {
        long long threads = (long long)N * 32;
        int blocks = (int)((threads + 255) / 256);
        k_attn_scalars<<<blocks, 256, 0, stream>>>(Wh, a_w, s1, s2, N);
    }

    k_alpha_max<<<(E + 255) / 256, 256, 0, stream>>>(src, dst, s1, s2, alpha_io, gmax, E);
    k_exp_sum<<<(E + 255) / 256, 256, 0, stream>>>(dst, alpha_io, alpha_sum, gmax, E);

    {
        long long threads = (long long)E * 32;
        int blocks = (int)((threads + 255) / 256);
        k_scatter<<<blocks, 256, 0, stream>>>(src, dst, Wh, alpha_sum, alpha_io, out_nodes, E);
    }

    k_elu<<<(n_out_elems + 255) / 256, 256, 0, stream>>>(out_nodes, n_out_elems);
}